// GNNdxdt_51900384804999
// MI455X (gfx1250) — compile-verified
//
#include <hip/hip_runtime.h>
#include <hip/hip_bf16.h>

typedef __attribute__((ext_vector_type(2))) float v2f;
typedef __attribute__((ext_vector_type(8))) float v8f;

__device__ __forceinline__ float silu_f(float x) {
    return x / (1.0f + __expf(-x));
}

// ---------------------------------------------------------------------------
// Degree / normalization kernels
// ---------------------------------------------------------------------------
__global__ void k_init_deg(float* __restrict__ deg, int N) {
    int i = blockIdx.x * blockDim.x + threadIdx.x;
    if (i < N) deg[i] = 1.0f;  // self-loop contributes 1 to in-degree
}

__global__ void k_edge_deg(const long long* __restrict__ dst,
                           float* __restrict__ deg, int E) {
    int e = blockIdx.x * blockDim.x + threadIdx.x;
    if (e < E) atomicAdd(&deg[(int)dst[e]], 1.0f);
}

// dinv = rsqrt(deg) (deg >= 1 always, self-loops);  s = x * dinv^2 (self-loop term)
__global__ void k_dinv_sinit(const float* __restrict__ x,
                             float* __restrict__ deg_dinv,
                             float* __restrict__ s, int N) {
    int i = blockIdx.x * blockDim.x + threadIdx.x;
    if (i < N) {
        float d = rsqrtf(deg_dinv[i]);
        deg_dinv[i] = d;
        s[i] = x[i] * d * d;
    }
}

// Layer-1 scalar scatter: s[dst] += x[src] * dinv[src] * dinv[dst]
__global__ void k_edge_scalar(const long long* __restrict__ src,
                              const long long* __restrict__ dst,
                              const float* __restrict__ x,
                              const float* __restrict__ dinv,
                              float* __restrict__ s, int E) {
    int e = blockIdx.x * blockDim.x + threadIdx.x;
    if (e < E) {
        int si = (int)src[e], di = (int)dst[e];
        atomicAdd(&s[di], x[si] * dinv[si] * dinv[di]);
    }
}

// h1[n,f] = silu(s[n] * W1[f] + b1[f])   (rank-1 expansion of GCN layer 1)
__global__ void k_layer1(const float* __restrict__ s,
                         const float* __restrict__ W1,
                         const float* __restrict__ b1,
                         float* __restrict__ h1, int N) {
    int i = blockIdx.x * blockDim.x + threadIdx.x;
    if (i < N * 64) {
        int n = i >> 6, f = i & 63;
        h1[i] = silu_f(s[n] * W1[f] + b1[f]);
    }
}

// ---------------------------------------------------------------------------
// WMMA GEMM:  C[N,64] = A[N,64] @ W[64,64]   (+ optional fused bias + SiLU)
// One wave computes one 16x16 output tile via 16x V_WMMA_F32_16X16X4_F32.
// B (weight) tile preloaded in 16 v2f registers one per K step.
// ---------------------------------------------------------------------------
template <int FUSE>
__global__ __launch_bounds__(256) void k_gemm64(const float* __restrict__ A,
                                                const float* __restrict__ W,
                                                const float* __restrict__ bias,
                                                float* __restrict__ C,
                                                int nTiles) {
    const int lane = threadIdx.x & 31;
    const int wv   = threadIdx.x >> 5;
    const int rowTile = blockIdx.y * 8 + wv;   // wave-uniform
    if (rowTile >= nTiles) return;             // whole-wave exit, EXEC stays full

    const int hi  = lane >> 4;       // half-wave select
    const int m   = lane & 15;       // A row / B,C column within tile
    const int kb  = hi * 2;          // K base within 4-wide K slab
    const int col = (blockIdx.x << 4) + m;

    // Preload B: for k-step k, VGPR pair holds W[4k+kb][col], W[4k+kb+1][col]
    v2f breg[16];
#pragma unroll
    for (int k = 0; k < 16; ++k) {
        int kr = k * 4 + kb;
        breg[k].x = W[kr * 64 + col];
        breg[k].y = W[(kr + 1) * 64 + col];
    }

    const float* arow = A + (rowTile * 16 + m) * 64 + kb;
    v8f acc = {};
#pragma unroll
    for (int k = 0; k < 16; ++k) {
        v2f a;
        a.x = arow[k * 4];
        a.y = arow[k * 4 + 1];
        acc = __builtin_amdgcn_wmma_f32_16x16x4_f32(
            /*neg_a=*/false, a, /*neg_b=*/false, breg[k],
            /*c_mod=*/(short)0, acc, /*reuse_a=*/false, /*reuse_b=*/false);
    }

    float bval = FUSE ? bias[col] : 0.0f;
    const int rbase = rowTile * 16 + hi * 8;   // C/D layout: VGPR r -> row r + 8*hi
#pragma unroll
    for (int r = 0; r < 8; ++r) {
        float v = acc[r];
        if (FUSE) v = silu_f(v + bval);
        C[(rbase + r) * 64 + col] = v;
    }
}

// acc2[n,f] = xw2[n,f] * dinv[n]^2   (layer-2 self-loop term)
__global__ void k_conv2_init(const float* __restrict__ xw,
                             const float* __restrict__ dinv,
                             float* __restrict__ acc, int N) {
    int i = blockIdx.x * blockDim.x + threadIdx.x;
    if (i < N * 64) {
        float d = dinv[i >> 6];
        acc[i] = xw[i] * d * d;
    }
}

// Layer-2 message scatter: 64 threads per edge (e uniform within each wave32).
// acc[dst,f] += xw2[src,f] * dinv[src] * dinv[dst]
__global__ void k_edge_vec(const long long* __restrict__ src,
                           const long long* __restrict__ dst,
                           const float* __restrict__ xw,
                           const float* __restrict__ dinv,
                           float* __restrict__ acc, int E) {
    unsigned gid = blockIdx.x * blockDim.x + threadIdx.x;
    int e = gid >> 6;
    int f = gid & 63;
    if (e < E) {
        int si = (int)src[e], di = (int)dst[e];
        float norm = dinv[si] * dinv[di];
        atomicAdd(&acc[di * 64 + f], xw[si * 64 + f] * norm);
    }
}

__global__ void k_bias_silu(float* __restrict__ a,
                            const float* __restrict__ b, int N) {
    int i = blockIdx.x * blockDim.x + threadIdx.x;
    if (i < N * 64) a[i] = silu_f(a[i] + b[i & 63]);
}

// out[n] = dot(h[n,:], W4) + b4   — wave32 per node, float2 lanes + shuffle tree
__global__ void k_out(const float* __restrict__ h,
                      const float* __restrict__ W4,
                      const float* __restrict__ b4,
                      float* __restrict__ out, int N) {
    int lane = threadIdx.x & 31;
    int wv   = threadIdx.x >> 5;
    int n    = blockIdx.x * 8 + wv;
    if (n >= N) return;
    float2 hv = ((const float2*)(h + n * 64))[lane];
    float2 wvv = ((const float2*)W4)[lane];
    float v = hv.x * wvv.x + hv.y * wvv.y;
#pragma unroll
    for (int off = 16; off > 0; off >>= 1) v += __shfl_down(v, off, 32);
    if (lane == 0) out[n] = v + b4[0];
}

// ---------------------------------------------------------------------------
extern "C" void kernel_launch(void* const* d_in, const int* in_sizes, int n_in,
                              void* d_out, int out_size, void* d_ws, size_t ws_size,
                              hipStream_t stream) {
    const float*     x   = (const float*)d_in[0];
    const long long* ei  = (const long long*)d_in[1];  // int64 [2,E]
    const float*     W1  = (const float*)d_in[2];
    const float*     b1  = (const float*)d_in[3];
    const float*     W2  = (const float*)d_in[4];
    const float*     b2  = (const float*)d_in[5];
    const float*     W3  = (const float*)d_in[6];
    const float*     b3  = (const float*)d_in[7];
    const float*     W4  = (const float*)d_in[8];
    const float*     b4  = (const float*)d_in[9];
    float* out = (float*)d_out;

    const int N = in_sizes[0];        // 100000 (multiple of 16)
    const int E = in_sizes[1] / 2;    // 1600000
    const long long* src = ei;
    const long long* dst = ei + E;

    float* ws   = (float*)d_ws;
    float* dinv = ws;                         // N floats (deg -> dinv in place)
    float* s    = ws + N;                     // N floats
    float* bufA = ws + 2 * (size_t)N;         // N*64: h1 -> acc2/h2
    float* bufB = bufA + (size_t)N * 64;      // N*64: xw2 -> h3

    const int T = 256;
    const int NF = N * 64;

    k_init_deg   <<<(N  + T - 1) / T, T, 0, stream>>>(dinv, N);
    k_edge_deg   <<<(E  + T - 1) / T, T, 0, stream>>>(dst, dinv, E);
    k_dinv_sinit <<<(N  + T - 1) / T, T, 0, stream>>>(x, dinv, s, N);
    k_edge_scalar<<<(E  + T - 1) / T, T, 0, stream>>>(src, dst, x, dinv, s, E);
    k_layer1     <<<(NF + T - 1) / T, T, 0, stream>>>(s, W1, b1, bufA, N);

    const int nTiles = (N + 15) / 16;
    dim3 gg(4, (nTiles + 7) / 8);
    k_gemm64<0><<<gg, 256, 0, stream>>>(bufA, W2, nullptr, bufB, nTiles);  // xw2

    k_conv2_init <<<(NF + T - 1) / T, T, 0, stream>>>(bufB, dinv, bufA, N);
    long long tot = (long long)E * 64;
    k_edge_vec   <<<(unsigned)((tot + T - 1) / T), T, 0, stream>>>(src, dst, bufB, dinv, bufA, E);
    k_bias_silu  <<<(NF + T - 1) / T, T, 0, stream>>>(bufA, b2, N);

    k_gemm64<1><<<gg, 256, 0, stream>>>(bufA, W3, b3, bufB, nTiles);       // h3 = silu(h2@W3+b3)

    k_out<<<(N + 7) / 8, 256, 0, stream>>>(bufB, W4, b4, out, N);
}